// GraphNet_79233556676742
// MI455X (gfx1250) — compile-verified
//
#include <hip/hip_runtime.h>
#include <hip/hip_bf16.h>

// ---------------------------------------------------------------------------
// GraphNet (3x EdgeConv + GAT) for gfx1250.
//
// Algebraic folding per edge-conv layer:
//   msg_e = h[src_e] @ (wn@wc_top) + ea_e @ (we@wc_bot) + (bn@wc_top+be@wc_bot+bc)
// so each layer = one N x 64 x 64 GEMM (fp32 WMMA 16x16x4) + an edge
// gather/affine/atomic-scatter pass. GAT head: N x 64 x 128 WMMA GEMM,
// segment-softmax via ordered-uint atomicMax + f32 atomics.
//
// B matrices are pre-packed into per-lane WMMA fragment order so the GEMM
// inner loop is 16 global_load_b64 followed by 16 back-to-back v_wmma ops.
// ---------------------------------------------------------------------------

typedef float v2f __attribute__((ext_vector_type(2)));
typedef float v8f __attribute__((ext_vector_type(8)));

#define NSLOPE 0.2f

// Packed-B element index for B[k][c] (k in 0..63, c in 0..ldout-1):
//   t = k>>2, half = (k>>1)&1, j = k&1, nt = c>>4, l15 = c&15, lane = half*16+l15
//   Bp[ ((nt*16 + t)*32 + lane)*2 + j ]
__device__ __forceinline__ int packedBIdx(int k, int c) {
    int t = k >> 2, half = (k >> 1) & 1, j = k & 1;
    int nt = c >> 4, l15 = c & 15;
    int lane = half * 16 + l15;
    return ((nt * 16 + t) * 32 + lane) * 2 + j;
}

// -------------------- generic fill --------------------
__global__ void fill_f32(float* __restrict__ p, float v, long count) {
    long i = (long)blockIdx.x * blockDim.x + threadIdx.x;
    if (i < count) p[i] = v;
}

// -------------------- pack a dense [64, ldout] B into fragment order --------
__global__ void pack_B(const float* __restrict__ B, float* __restrict__ Bp, int ldout) {
    int idx = blockIdx.x * blockDim.x + threadIdx.x;   // (nt*16 + t)*32 + lane
    int total = (ldout >> 4) * 16 * 32;
    if (idx >= total) return;
    int lane = idx & 31;
    int t    = (idx >> 5) & 15;
    int nt   = idx >> 9;
    int half = lane >> 4, l15 = lane & 15;
    int k   = t * 4 + half * 2;
    int col = nt * 16 + l15;
    Bp[idx * 2 + 0] = B[(k + 0) * ldout + col];
    Bp[idx * 2 + 1] = B[(k + 1) * ldout + col];
}

// -------------------- weight folding (tiny), writes packed Wf ---------------
// Wfp  = pack(wn @ wc[0:64,:])   (64x64, fragment order)
// Wef  = we @ wc[64:128,:]       (5x64, dense)
// btot = bn@wc_top + be@wc_bot + bc   (64)
__global__ void fold_layer(const float* __restrict__ wn, const float* __restrict__ bn,
                           const float* __restrict__ we, const float* __restrict__ be,
                           const float* __restrict__ wc, const float* __restrict__ bc,
                           float* __restrict__ Wfp, float* __restrict__ Wef,
                           float* __restrict__ btot) {
    int idx = blockIdx.x * blockDim.x + threadIdx.x;
    if (idx < 4096) {
        int r = idx >> 6, c = idx & 63;       // r = K row of B, c = output col
        float s = 0.f;
        for (int k = 0; k < 64; ++k) s += wn[r * 64 + k] * wc[k * 64 + c];
        Wfp[packedBIdx(r, c)] = s;
    } else if (idx < 4096 + 320) {
        int j = idx - 4096;
        int r = j >> 6, c = j & 63;
        float s = 0.f;
        for (int k = 0; k < 64; ++k) s += we[r * 64 + k] * wc[(64 + k) * 64 + c];
        Wef[j] = s;
    } else if (idx < 4096 + 320 + 64) {
        int c = idx - 4096 - 320;
        float s = bc[c];
        for (int k = 0; k < 64; ++k)
            s += bn[k] * wc[k * 64 + c] + be[k] * wc[(64 + k) * 64 + c];
        btot[c] = s;
    }
}

// -------------------- fp32 WMMA GEMM: D[n,ldout] = (relu?)A[n,64] @ B -------
// B given pre-packed in fragment order. One wave per 16-row strip; A fragments
// cached in registers across N-tiles; per tile: 16x b64 loads then 16 WMMAs.
__global__ void __launch_bounds__(256)
gemm64_wmma(const float* __restrict__ A, const float* __restrict__ Bp,
            float* __restrict__ D, int n, int ldout, int relu_in) {
    const int wavesPerBlk = blockDim.x >> 5;
    const int mt = blockIdx.x * wavesPerBlk + (threadIdx.x >> 5);
    if (mt * 16 >= n) return;                 // wave-uniform: EXEC stays all-1s
    const int lane = threadIdx.x & 31;
    const int half = lane >> 4;
    const int l15  = lane & 15;
    const int row  = mt * 16 + l15;

    // A fragments: lane (l15,half) holds A[row][t*4 + half*2 + {0,1}]
    v2f aF[16];
    const float* Ar = A + (long)row * 64;
#pragma unroll
    for (int t = 0; t < 16; ++t)
        aF[t] = *(const v2f*)(Ar + t * 4 + half * 2);
    if (relu_in) {
#pragma unroll
        for (int t = 0; t < 16; ++t) {
            aF[t].x = fmaxf(aF[t].x, 0.f);
            aF[t].y = fmaxf(aF[t].y, 0.f);
        }
    }

    const int ntiles = ldout >> 4;
    for (int nt = 0; nt < ntiles; ++nt) {
        const float* Bnt = Bp + nt * (16 * 64);
        v2f bF[16];
#pragma unroll
        for (int t = 0; t < 16; ++t)
            bF[t] = *(const v2f*)(Bnt + (t * 32 + lane) * 2);   // one b64 each
        v8f acc = {};
#pragma unroll
        for (int t = 0; t < 16; ++t)
            acc = __builtin_amdgcn_wmma_f32_16x16x4_f32(
                false, aF[t], false, bF[t], (short)0, acc, false, false);
        float* Dp = D + (long)(mt * 16 + half * 8) * ldout + nt * 16 + l15;
#pragma unroll
        for (int i = 0; i < 8; ++i) Dp[(long)i * ldout] = acc[i];
    }
}

// -------------------- edge conv: acc[dst] += p[src] + ea@Wef + btot --------
__global__ void edgeconv_scatter(const float* __restrict__ p,
                                 const int* __restrict__ src, const int* __restrict__ dst,
                                 const float* __restrict__ ea,
                                 const float* __restrict__ Wef,
                                 const float* __restrict__ btot,
                                 float* __restrict__ acc, int E_) {
    long idx = (long)blockIdx.x * blockDim.x + threadIdx.x;
    if (idx >= (long)E_ * 64) return;
    int e = (int)(idx >> 6);
    int c = (int)(idx & 63);
    int s = src[e], d = dst[e];
    float v = p[(long)s * 64 + c] + btot[c];
#pragma unroll
    for (int k = 0; k < 5; ++k) v += ea[(long)e * 5 + k] * Wef[k * 64 + c];
    atomicAdd(&acc[(long)d * 64 + c], v);
}

// -------------------- GAT --------------------
__global__ void gat_scores(const float* __restrict__ g,
                           const float* __restrict__ att_s, const float* __restrict__ att_d,
                           float* __restrict__ a_src, float* __restrict__ a_dst, int n) {
    int idx = blockIdx.x * blockDim.x + threadIdx.x;
    if (idx >= n * 2) return;
    int node = idx >> 1, h = idx & 1;
    const float* gr = g + (long)node * 128 + h * 64;
    float ss = 0.f, sd = 0.f;
    for (int c = 0; c < 64; ++c) {
        float v = gr[c];
        ss += v * att_s[h * 64 + c];
        sd += v * att_d[h * 64 + c];
    }
    a_src[idx] = ss;
    a_dst[idx] = sd;
}

__device__ __forceinline__ unsigned f2key(float f) {
    unsigned b = __float_as_uint(f);
    return (b & 0x80000000u) ? ~b : (b | 0x80000000u);
}
__device__ __forceinline__ float key2f(unsigned k) {
    return __uint_as_float((k & 0x80000000u) ? (k ^ 0x80000000u) : ~k);
}

__global__ void gat_alpha_max(const int* __restrict__ src, const int* __restrict__ dst,
                              const float* __restrict__ a_src, const float* __restrict__ a_dst,
                              float* __restrict__ alpha, unsigned* __restrict__ mkey, int E_) {
    int idx = blockIdx.x * blockDim.x + threadIdx.x;
    if (idx >= E_ * 2) return;
    int e = idx >> 1, h = idx & 1;
    float a = a_src[src[e] * 2 + h] + a_dst[dst[e] * 2 + h];
    a = (a > 0.f) ? a : NSLOPE * a;              // leaky_relu
    alpha[idx] = a;
    atomicMax(&mkey[dst[e] * 2 + h], f2key(a));
}

__global__ void gat_exp_denom(const int* __restrict__ dst,
                              float* __restrict__ alpha, const unsigned* __restrict__ mkey,
                              float* __restrict__ denom, int E_) {
    int idx = blockIdx.x * blockDim.x + threadIdx.x;
    if (idx >= E_ * 2) return;
    int e = idx >> 1, h = idx & 1;
    float m = key2f(mkey[dst[e] * 2 + h]);
    float v = expf(alpha[idx] - m);
    alpha[idx] = v;
    atomicAdd(&denom[dst[e] * 2 + h], v);
}

__global__ void gat_scatter(const int* __restrict__ src, const int* __restrict__ dst,
                            const float* __restrict__ alpha, const float* __restrict__ denom,
                            const float* __restrict__ g, float* __restrict__ gacc, int E_) {
    long idx = (long)blockIdx.x * blockDim.x + threadIdx.x;
    if (idx >= (long)E_ * 128) return;
    int e = (int)(idx >> 7);
    int t = (int)(idx & 127);
    int h = t >> 6;
    int d = dst[e];
    float coef = alpha[e * 2 + h] / (denom[d * 2 + h] + 1e-16f);
    atomicAdd(&gacc[(long)d * 128 + t], coef * g[(long)src[e] * 128 + t]);
}

__global__ void gat_final(const float* __restrict__ gacc, const float* __restrict__ bgat,
                          float* __restrict__ out, int n) {
    long idx = (long)blockIdx.x * blockDim.x + threadIdx.x;
    if (idx >= (long)n * 64) return;
    int node = (int)(idx >> 6);
    int c = (int)(idx & 63);
    out[idx] = 0.5f * (gacc[(long)node * 128 + c] + gacc[(long)node * 128 + 64 + c]) + bgat[c];
}

// ---------------------------------------------------------------------------
static inline int gdiv(long a, int b) { return (int)((a + b - 1) / b); }

extern "C" void kernel_launch(void* const* d_in, const int* in_sizes, int n_in,
                              void* d_out, int out_size, void* d_ws, size_t ws_size,
                              hipStream_t stream) {
    const float* x  = (const float*)d_in[0];
    const int*   ei = (const int*)d_in[1];
    const float* ea = (const float*)d_in[2];
    const int n  = in_sizes[0] / 64;
    const int E_ = in_sizes[1] / 2;
    const int* src = ei;
    const int* dst = ei + E_;

    const float *wn[3], *bn[3], *we[3], *be[3], *wc[3], *bc[3];
    for (int i = 0; i < 3; ++i) {
        int b = 3 + i * 6;
        wn[i] = (const float*)d_in[b + 0];
        bn[i] = (const float*)d_in[b + 1];
        we[i] = (const float*)d_in[b + 2];
        be[i] = (const float*)d_in[b + 3];
        wc[i] = (const float*)d_in[b + 4];
        bc[i] = (const float*)d_in[b + 5];
    }
    const float* wgat  = (const float*)d_in[21];
    const float* att_s = (const float*)d_in[22];
    const float* att_d = (const float*)d_in[23];
    const float* bgat  = (const float*)d_in[24];

    // workspace layout (floats); total ~34.4M floats (~138 MB)
    float* ws = (float*)d_ws;
    size_t off = 0;
    float* WFP   = ws + off; off += 3 * 4096;        // packed folded node weights
    float* WGATP = ws + off; off += 8192;            // packed w_gat
    float* WEF   = ws + off; off += 3 * 320;
    float* BTOT  = ws + off; off += 3 * 64;
    float* ASRC  = ws + off; off += (size_t)n * 2;
    float* ADST  = ws + off; off += (size_t)n * 2;
    float* DEN   = ws + off; off += (size_t)n * 2;
    unsigned* MKEY = (unsigned*)(ws + off); off += (size_t)n * 2;
    float* ALPHA = ws + off; off += (size_t)E_ * 2;
    float* P     = ws + off; off += (size_t)n * 64;
    float* ACCA  = ws + off; off += (size_t)n * 64;
    float* ACCB  = ws + off; off += (size_t)n * 64;
    float* G     = ws + off; off += (size_t)n * 128;
    float* GACC  = P;  // aliases [P, ACCB): both dead by the time GACC is used

    const int B256 = 256;
    const int gemmBlocks = gdiv((long)(n / 16), 8);           // 8 waves/block

    // fold + pack weights (tiny)
    for (int l = 0; l < 3; ++l)
        fold_layer<<<gdiv(4480, B256), B256, 0, stream>>>(
            wn[l], bn[l], we[l], be[l], wc[l], bc[l],
            WFP + l * 4096, WEF + l * 320, BTOT + l * 64);
    pack_B<<<gdiv(4096, B256), B256, 0, stream>>>(wgat, WGATP, 128);

    // ---- layer 1: x -> ACCA ----
    gemm64_wmma<<<gemmBlocks, B256, 0, stream>>>(x, WFP + 0 * 4096, P, n, 64, 0);
    fill_f32<<<gdiv((long)n * 64, B256), B256, 0, stream>>>(ACCA, 0.f, (long)n * 64);
    edgeconv_scatter<<<gdiv((long)E_ * 64, B256), B256, 0, stream>>>(
        P, src, dst, ea, WEF + 0 * 320, BTOT + 0 * 64, ACCA, E_);

    // ---- layer 2: relu(ACCA) -> ACCB ----
    gemm64_wmma<<<gemmBlocks, B256, 0, stream>>>(ACCA, WFP + 1 * 4096, P, n, 64, 1);
    fill_f32<<<gdiv((long)n * 64, B256), B256, 0, stream>>>(ACCB, 0.f, (long)n * 64);
    edgeconv_scatter<<<gdiv((long)E_ * 64, B256), B256, 0, stream>>>(
        P, src, dst, ea, WEF + 1 * 320, BTOT + 1 * 64, ACCB, E_);

    // ---- layer 3: relu(ACCB) -> ACCA ----
    gemm64_wmma<<<gemmBlocks, B256, 0, stream>>>(ACCB, WFP + 2 * 4096, P, n, 64, 1);
    fill_f32<<<gdiv((long)n * 64, B256), B256, 0, stream>>>(ACCA, 0.f, (long)n * 64);
    edgeconv_scatter<<<gdiv((long)E_ * 64, B256), B256, 0, stream>>>(
        P, src, dst, ea, WEF + 2 * 320, BTOT + 2 * 64, ACCA, E_);

    // ---- GAT head ----
    gemm64_wmma<<<gemmBlocks, B256, 0, stream>>>(ACCA, WGATP, G, n, 128, 1);

    fill_f32<<<gdiv((long)n * 128, B256), B256, 0, stream>>>(GACC, 0.f, (long)n * 128);
    fill_f32<<<gdiv((long)n * 2, B256), B256, 0, stream>>>(DEN, 0.f, (long)n * 2);
    fill_f32<<<gdiv((long)n * 2, B256), B256, 0, stream>>>((float*)MKEY, 0.f, (long)n * 2); // key 0 == -inf

    gat_scores<<<gdiv((long)n * 2, B256), B256, 0, stream>>>(G, att_s, att_d, ASRC, ADST, n);
    gat_alpha_max<<<gdiv((long)E_ * 2, B256), B256, 0, stream>>>(src, dst, ASRC, ADST, ALPHA, MKEY, E_);
    gat_exp_denom<<<gdiv((long)E_ * 2, B256), B256, 0, stream>>>(dst, ALPHA, MKEY, DEN, E_);
    gat_scatter<<<gdiv((long)E_ * 128, B256), B256, 0, stream>>>(src, dst, ALPHA, DEN, G, GACC, E_);
    gat_final<<<gdiv((long)n * 64, B256), B256, 0, stream>>>(GACC, bgat, (float*)d_out, n);
}